// ShapleyFusion_62534723829964
// MI455X (gfx1250) — compile-verified
//
#include <hip/hip_runtime.h>

// ShapleyFusion for MI455X (gfx1250, wave32).
//
// Bandwidth-bound kernel (~50 MB HBM traffic, ~75 MFLOP): read feats once,
// second pass hits L2 (42 MB << 192 MB). Dot-product stage s_l = <x_l, w>
// runs on the matrix pipe via V_WMMA_F32_16X16X4_F32:
//   A (16x4)  = w[c..c+3] replicated in every row  -> all D rows identical
//   B (4x16)  = feats[l, c..c+3, pixel-tile]       -> D[., N] = s over tile
// Two 16-pixel tiles per wave => after K-loop, acc VGPR0 holds this lane's
// own pixel score (lanes 0-15 from tile A, lanes 16-31 from tile B).
//
// Subset lattice: scores are linear, so f_m = relu(sum_{l in m} s_l + b);
// Shapley coefficients depend only on popcount:
//   wt(2)=1!*3!/5! = 0.05, wt(3)=2!*2!/5! = 1/30, wt(4)=3!*1!/5! = 0.05,
//   wt(5)=4!*0!/5! = 0.2, singleton coef = -(L-2)/L = -0.6.

typedef __attribute__((ext_vector_type(2))) float v2f;
typedef __attribute__((ext_vector_type(8))) float v8f;

constexpr int kL = 5;
constexpr int kC = 64;
constexpr int kPixPerBlock = 256;   // 8 waves * 32 pixels

__global__ __launch_bounds__(256)
void shapley_fusion_kernel(const float* __restrict__ feats,
                           const float* __restrict__ w,
                           const float* __restrict__ bptr,
                           float* __restrict__ out,
                           int N) {
  const int lane = threadIdx.x & 31;
  const int wv   = threadIdx.x >> 5;
  const int half = lane >> 4;            // 0: lanes 0-15, 1: lanes 16-31
  const int l16  = lane & 15;
  const int n0   = blockIdx.x * kPixPerBlock + wv * 32;  // wave's 32 pixels
  const int n    = n0 + lane;            // this lane's pixel (phases B/C)
  const int koff = 2 * half;             // K offset of this half-wave's frags

  const float bias = bptr[0];

  // ---------------- Phase A: per-view scores via WMMA ----------------
  float s[kL];
#pragma unroll
  for (int l = 0; l < kL; ++l) {
    v8f accA = {};                       // pixels n0 .. n0+15
    v8f accB = {};                       // pixels n0+16 .. n0+31
    const float* fb = feats + (size_t)l * kC * N;
    for (int c = 0; c < kC; c += 4) {
      // A fragment: w replicated across all 16 rows.
      // Layout: VGPR0 = K(0|2) per half-wave, VGPR1 = K(1|3).
      v2f a;
      a.x = w[c + koff];
      a.y = w[c + koff + 1];
      // B fragments: column = pixel (lane), rows = channels c..c+3.
      const float* pA = fb + (size_t)(c + koff) * N + (n0 + l16);
      v2f bA; bA.x = pA[0]; bA.y = pA[N];
      const float* pB = pA + 16;
      v2f bB; bB.x = pB[0]; bB.y = pB[N];
      accA = __builtin_amdgcn_wmma_f32_16x16x4_f32(
          false, a, false, bA, (short)0, accA, false, false);
      accB = __builtin_amdgcn_wmma_f32_16x16x4_f32(
          false, a, false, bB, (short)0, accB, false, false);
    }
    // D rows all equal; VGPR0: lanes 0-15 -> tileA col lane,
    // lanes 16-31 -> tileB col lane-16  ==> pixel n0+lane either way.
    s[l] = half ? accB[0] : accA[0];
  }

  // ---------------- Phase B: subset lattice + Shapley + softmax ----------------
  float sh[kL] = {0.f, 0.f, 0.f, 0.f, 0.f};
#pragma unroll
  for (int m = 1; m < (1 << kL); ++m) {
    float sum = bias;
#pragma unroll
    for (int l = 0; l < kL; ++l)
      if (m & (1 << l)) sum += s[l];
    const float f = fmaxf(sum, 0.0f);
    const int pc = __popc(m);            // folds to a constant per unrolled m
    const float wgt = (pc == 1) ? -0.6f
                    : (pc == 2) ? 0.05f
                    : (pc == 3) ? (1.0f / 30.0f)
                    : (pc == 4) ? 0.05f
                                : 0.2f;
#pragma unroll
    for (int l = 0; l < kL; ++l)
      if (m & (1 << l)) sh[l] += wgt * f;
  }

  float mx = sh[0];
#pragma unroll
  for (int l = 1; l < kL; ++l) mx = fmaxf(mx, sh[l]);
  float attn[kL];
  float tot = 0.f;
#pragma unroll
  for (int l = 0; l < kL; ++l) {
    attn[l] = __expf(sh[l] - mx);
    tot += attn[l];
  }
  const float inv = 1.0f / tot;
#pragma unroll
  for (int l = 0; l < kL; ++l) attn[l] *= inv;

  // ---------------- Phase C: fused output (reload feats from L2) ----------------
  const float* fp = feats + n;
  for (int c = 0; c < kC; ++c) {
    float acc = 0.f;
#pragma unroll
    for (int l = 0; l < kL; ++l)
      acc = fmaf(attn[l], fp[((size_t)l * kC + c) * N], acc);
    out[(size_t)c * N + n] = acc;        // output layout [C, H*W]
  }
}

extern "C" void kernel_launch(void* const* d_in, const int* in_sizes, int n_in,
                              void* d_out, int out_size, void* d_ws, size_t ws_size,
                              hipStream_t stream) {
  const float* feats = (const float*)d_in[0];
  const float* w     = (const float*)d_in[1];
  const float* b     = (const float*)d_in[2];
  float* out = (float*)d_out;
  const int N = in_sizes[0] / (kL * kC);           // H*W = 32768
  const int blocks = N / kPixPerBlock;             // 128 full blocks
  shapley_fusion_kernel<<<blocks, 256, 0, stream>>>(feats, w, b, out, N);
}